// MambaMine_25108378812675
// MI455X (gfx1250) — compile-verified
//
#include <hip/hip_runtime.h>

// Mamba-2 SSD forward for gfx1250 (MI455X), fp32 via V_WMMA_F32_16X16X4_F32.
// Data movement: double-buffered GLOBAL_LOAD_ASYNC_TO_LDS_B128 staging
// (ASYNCcnt) so global fills overlap the WMMA pipe.
// Dims: B=2 L=2048 D_MODEL=1024 D_SSM=2048 H=32 P=64 N=128 CHUNK=256 nc=8
// CONV_DIM=2304 D_IN_PROJ=4384
// Workspace: 45,481,984 floats (~182 MB).

typedef float v2f __attribute__((ext_vector_type(2)));
typedef float v8f __attribute__((ext_vector_type(8)));

#define WMMA_F32(a, b, c) \
  __builtin_amdgcn_wmma_f32_16x16x4_f32(false, (a), false, (b), (short)0, (c), false, false)

// Async global->LDS copy (CDNA5 VGLOBAL op, tracked by ASYNCcnt).
// LDS operand is the 32-bit LDS offset (low half of the generic pointer:
// shared aperture lives in the high 32 bits per the flat-addressing rules).
static __device__ __forceinline__ void async_b128(const void* lds, const void* gaddr) {
  asm volatile("global_load_async_to_lds_b128 %0, %1, off"
               :: "v"((unsigned)(unsigned long long)lds),
                  "v"((unsigned long long)gaddr)
               : "memory");
}
static __device__ __forceinline__ void async_wait0() {
  asm volatile("s_wait_asynccnt 0" ::: "memory");
}

// ---------------------------------------------------------------------------
// Generic batched GEMM: C[m,n] = sum_k A[m*lda+k] * B[n*ldb+k]   (C = A * B^T)
// BM=128, BN=32, KT=32. Requires M%128==0, N%32==0, K%32==0.
// grid = (M/128, N/32, nbatch), block = 256 (8 waves).
// Double-buffered async LDS staging: stage s+1 fills while stage s computes.
// ---------------------------------------------------------------------------
__global__ __launch_bounds__(256) void gemm_nt_f32(
    const float* __restrict__ A, int lda, long long strideA,
    const float* __restrict__ B, int ldb, long long strideB,
    float* __restrict__ C, int ldc, long long strideC, int K) {
  __shared__ float sA[2][128][36];  // pad 36: conflict-free column (K) reads
  __shared__ float sB[2][32][36];
  const int bm = blockIdx.x * 128, bn = blockIdx.y * 32;
  A += (long long)blockIdx.z * strideA;
  B += (long long)blockIdx.z * strideB;
  C += (long long)blockIdx.z * strideC;
  const int tid = threadIdx.x;
  const int w = tid >> 5, lane = tid & 31;
  const int wm = (w & 3) * 32;   // wave M offset within block tile
  const int wn = (w >> 2) * 16;  // wave N offset
  const int r = lane & 15, ks = (lane >> 4) * 2;

  // per-thread staging coordinates (4 float4 of A, 1 float4 of B per stage)
  const int arow0 = tid >> 3, ac4 = (tid & 7) * 4;  // + i*32 rows
  const int brow = tid >> 3, bc4 = (tid & 7) * 4;

  auto issue_stage = [&](int buf, int k0) {
    for (int i = 0; i < 4; ++i) {
      const int row = arow0 + i * 32;
      async_b128(&sA[buf][row][ac4],
                 A + (long long)(bm + row) * lda + (k0 + ac4));
    }
    async_b128(&sB[buf][brow][bc4],
               B + (long long)(bn + brow) * ldb + (k0 + bc4));
  };

  v8f acc0 = {}, acc1 = {};
  const int nstage = K >> 5;
  issue_stage(0, 0);
  async_wait0();
  __syncthreads();
  for (int s = 0; s < nstage; ++s) {
    const int buf = s & 1;
    if (s + 1 < nstage) issue_stage(buf ^ 1, (s + 1) * 32);  // overlap w/ WMMA
    for (int kk = 0; kk < 8; ++kk) {
      const int k = kk * 4 + ks;  // VGPR0=K, VGPR1=K+1 (lanes>=16: K+2/K+3)
      const v2f bf = *(const v2f*)&sB[buf][wn + r][k];
      const v2f a0 = *(const v2f*)&sA[buf][wm + r][k];
      const v2f a1 = *(const v2f*)&sA[buf][wm + 16 + r][k];
      acc0 = WMMA_F32(a0, bf, acc0);
      acc1 = WMMA_F32(a1, bf, acc1);
    }
    if (s + 1 < nstage) async_wait0();
    __syncthreads();
  }
  const int hi8 = (lane >> 4) * 8;
  for (int i = 0; i < 8; ++i) {
    C[(long long)(bm + wm + hi8 + i) * ldc + (bn + wn + r)] = acc0[i];
    C[(long long)(bm + wm + 16 + hi8 + i) * ldc + (bn + wn + r)] = acc1[i];
  }
}

// ---------------------------------------------------------------------------
// Causal depthwise conv1d (K=4) + bias + SiLU over xBC slice of zxbcdt.
// one thread per (b,l,c); id = (b*2048 + l)*2304 + c
// ---------------------------------------------------------------------------
__global__ __launch_bounds__(256) void conv_silu(
    const float* __restrict__ zxbcdt, const float* __restrict__ cw,
    const float* __restrict__ cb, float* __restrict__ xBC) {
  const long long id = (long long)blockIdx.x * 256 + threadIdx.x;
  const int cc = (int)(id % 2304);
  const long long t = id / 2304;
  const int l = (int)(t % 2048);
  const int b = (int)(t / 2048);
  float acc = cb[cc];
  for (int k = 0; k < 4; ++k) {
    const int ls = l - 3 + k;
    if (ls >= 0)
      acc += zxbcdt[((long long)(b * 2048 + ls)) * 4384 + 2048 + cc] * cw[cc * 4 + k];
  }
  xBC[id] = acc / (1.f + __expf(-acc));  // silu
}

// ---------------------------------------------------------------------------
// dt = softplus(dt_raw + bias); cum = cumsum(dt * A) per chunk.
// 512 threads total, one per (b,chunk,h), serial over 256 positions.
// ---------------------------------------------------------------------------
__global__ void dt_cumsum(const float* __restrict__ zxbcdt,
                          const float* __restrict__ dt_bias,
                          const float* __restrict__ A_log,
                          float* __restrict__ dtws, float* __restrict__ cumws) {
  const int g = blockIdx.x * blockDim.x + threadIdx.x;
  if (g >= 512) return;
  const int h = g & 31, c = (g >> 5) & 7, b = g >> 8;
  const float Ah = -__expf(A_log[h]);
  const float bias = dt_bias[h];
  float run = 0.f;
  for (int l = 0; l < 256; ++l) {
    const long long row = (long long)b * 2048 + c * 256 + l;
    const float raw = zxbcdt[row * 4384 + 4352 + h] + bias;
    const float dtv = (raw > 20.f) ? raw : log1pf(__expf(raw));
    run += dtv * Ah;
    dtws[row * 32 + h] = dtv;
    cumws[row * 32 + h] = run;
  }
}

// ---------------------------------------------------------------------------
// Intra-chunk: Y = M @ x  with M[l,s] = CB[l,s]*exp(cum_l-cum_s)*dt_s (s<=l),
// plus chunk state S[n,p] = sum_l B[l,n]*exp(cum_last-cum_l)*dt_l*x[l,p].
// One block per (h, chunk, b). x_h staged to LDS via async copies (64 KB).
// ---------------------------------------------------------------------------
__global__ __launch_bounds__(256) void ssd_intra(
    const float* __restrict__ xBC, const float* __restrict__ CBws,
    const float* __restrict__ cumws, const float* __restrict__ dtws,
    const float* __restrict__ Dvec, float* __restrict__ yws,
    float* __restrict__ Sws) {
  __shared__ float xh[256][64];
  const int h = blockIdx.x, c = blockIdx.y, b = blockIdx.z;
  const long long rowbase = (long long)b * 2048 + c * 256;
  const int tid = threadIdx.x;
  {  // async stage of x_h tile: 16 B128s per thread (ASYNCcnt max 63, ok)
    const int row0 = tid >> 4, c4 = (tid & 15) * 4;
    for (int i = 0; i < 16; ++i) {
      const int row = row0 + i * 16;
      async_b128(&xh[row][c4],
                 xBC + (rowbase + row) * 2304 + h * 64 + c4);
    }
    async_wait0();
  }
  __syncthreads();
  const int w = tid >> 5, lane = tid & 31;
  const int r = lane & 15, ks = (lane >> 4) * 2;
  const float* cumg = cumws + rowbase * 32 + h;  // cum[l] at cumg[l*32]
  const float* dtg = dtws + rowbase * 32 + h;
  const float clast = cumg[255 * 32];
  const float Dv = Dvec[h];
  const float* CBg = CBws + ((long long)b * 8 + c) * 65536;
  const int hi8 = (lane >> 4) * 8;

  // ---- Y_intra: triangular; wave w takes l-tiles {w, 15-w} (balanced) ----
  for (int tt = 0; tt < 2; ++tt) {
    const int t = tt ? (15 - w) : w;
    const int l0 = t * 16;
    const int lrow = l0 + r;
    const float cl = cumg[lrow * 32];
    v8f acc[4] = {};
    const int nk = 4 * t + 4;  // only s-tiles with s <= l0+15
    for (int kk = 0; kk < nk; ++kk) {
      const int s0 = kk * 4 + ks;
      v2f a;
      {
        const float m0 = CBg[lrow * 256 + s0] * __expf(cl - cumg[s0 * 32]) * dtg[s0 * 32];
        const float m1 = CBg[lrow * 256 + s0 + 1] * __expf(cl - cumg[(s0 + 1) * 32]) * dtg[(s0 + 1) * 32];
        a.x = (s0 <= lrow) ? m0 : 0.f;
        a.y = (s0 + 1 <= lrow) ? m1 : 0.f;
      }
      for (int pt = 0; pt < 4; ++pt) {  // A-frag (exp cost) reused over 4 N-tiles
        v2f bf;
        bf.x = xh[s0][pt * 16 + r];
        bf.y = xh[s0 + 1][pt * 16 + r];
        acc[pt] = WMMA_F32(a, bf, acc[pt]);
      }
    }
    for (int i = 0; i < 8; ++i) {
      const int m = l0 + hi8 + i;
      for (int pt = 0; pt < 4; ++pt) {
        const int p = pt * 16 + r;
        yws[(rowbase + m) * 2048 + h * 64 + p] = acc[pt][i] + Dv * xh[m][p];
      }
    }
  }

  // ---- chunk state: S = B^T @ (w .* x), wave w -> n-tile row w*16 ----
  const float* Bg = xBC + rowbase * 2304 + 2048;
  const int n0 = w * 16;
  v8f sacc[4] = {};
  for (int kk = 0; kk < 64; ++kk) {
    const int s0 = kk * 4 + ks;
    v2f a;
    a.x = Bg[(long long)s0 * 2304 + n0 + r];
    a.y = Bg[(long long)(s0 + 1) * 2304 + n0 + r];
    const float w0 = __expf(clast - cumg[s0 * 32]) * dtg[s0 * 32];
    const float w1 = __expf(clast - cumg[(s0 + 1) * 32]) * dtg[(s0 + 1) * 32];
    for (int pt = 0; pt < 4; ++pt) {
      v2f bf;
      bf.x = w0 * xh[s0][pt * 16 + r];
      bf.y = w1 * xh[s0 + 1][pt * 16 + r];
      sacc[pt] = WMMA_F32(a, bf, sacc[pt]);
    }
  }
  float* Sg = Sws + (((long long)b * 8 + c) * 32 + h) * 8192;
  for (int i = 0; i < 8; ++i)
    for (int pt = 0; pt < 4; ++pt)
      Sg[(n0 + hi8 + i) * 64 + pt * 16 + r] = sacc[pt][i];
}

// ---------------------------------------------------------------------------
// Inter-chunk recurrence over nc=8 chunks (emit state entering each chunk).
// One thread per (b,h,n,p) = 524288 threads.
// ---------------------------------------------------------------------------
__global__ void ssd_scan(const float* __restrict__ Sws,
                         const float* __restrict__ cumws,
                         float* __restrict__ prevws) {
  const int id = blockIdx.x * blockDim.x + threadIdx.x;
  const int p = id & 63, n = (id >> 6) & 127, h = (id >> 13) & 31, b = id >> 18;
  float carry = 0.f;
  for (int c = 0; c < 8; ++c) {
    const long long off = (((long long)b * 8 + c) * 32 + h) * 8192 + n * 64 + p;
    prevws[off] = carry;
    const float decay = __expf(cumws[((long long)(b * 2048 + c * 256 + 255)) * 32 + h]);
    carry = carry * decay + Sws[off];
  }
}

// ---------------------------------------------------------------------------
// Y_inter[l,p] = exp(cum_l) * sum_n C[l,n] * prev[n,p]; accumulate into y.
// One block per (h, chunk, b). M=256, N=64, K=128.
// ---------------------------------------------------------------------------
__global__ __launch_bounds__(256) void ssd_inter(
    const float* __restrict__ xBC, const float* __restrict__ prevws,
    const float* __restrict__ cumws, float* __restrict__ yws) {
  const int h = blockIdx.x, c = blockIdx.y, b = blockIdx.z;
  const long long rowbase = (long long)b * 2048 + c * 256;
  const int tid = threadIdx.x, w = tid >> 5, lane = tid & 31;
  const int r = lane & 15, ks = (lane >> 4) * 2;
  const float* Cg = xBC + rowbase * 2304 + 2176;
  const float* Pg = prevws + (((long long)b * 8 + c) * 32 + h) * 8192;
  const float* cumg = cumws + rowbase * 32 + h;
  for (int tt = 0; tt < 2; ++tt) {
    const int l0 = (w + tt * 8) * 16;
    v8f acc[4] = {};
    for (int kk = 0; kk < 32; ++kk) {
      const int n = kk * 4 + ks;
      const v2f a = *(const v2f*)(Cg + (long long)(l0 + r) * 2304 + n);
      for (int pt = 0; pt < 4; ++pt) {
        v2f bf;
        bf.x = Pg[n * 64 + pt * 16 + r];
        bf.y = Pg[(n + 1) * 64 + pt * 16 + r];
        acc[pt] = WMMA_F32(a, bf, acc[pt]);
      }
    }
    const int hi8 = (lane >> 4) * 8;
    for (int i = 0; i < 8; ++i) {
      const int m = l0 + hi8 + i;
      const float e = __expf(cumg[m * 32]);
      for (int pt = 0; pt < 4; ++pt) {
        const int p = pt * 16 + r;
        yws[(rowbase + m) * 2048 + h * 64 + p] += e * acc[pt][i];
      }
    }
  }
}

// ---------------------------------------------------------------------------
// Gated RMSNorm in-place: yg = y*silu(z); y = yg*rsqrt(mean(yg^2)+eps)*norm_w
// One block per token row (4096 rows), 8 elements/thread.
// ---------------------------------------------------------------------------
__global__ __launch_bounds__(256) void gated_rmsnorm(
    float* __restrict__ yws, const float* __restrict__ zxbcdt,
    const float* __restrict__ normw) {
  __shared__ float red[256];
  const long long row = blockIdx.x;
  const int tid = threadIdx.x;
  float g[8];
  float ss = 0.f;
  for (int i = 0; i < 8; ++i) {
    const int e = tid + i * 256;
    const float z = zxbcdt[row * 4384 + e];
    const float gv = yws[row * 2048 + e] * (z / (1.f + __expf(-z)));
    g[i] = gv;
    ss += gv * gv;
  }
  red[tid] = ss;
  __syncthreads();
  for (int s = 128; s > 0; s >>= 1) {
    if (tid < s) red[tid] += red[tid + s];
    __syncthreads();
  }
  const float scale = rsqrtf(red[0] * (1.f / 2048.f) + 1e-5f);
  for (int i = 0; i < 8; ++i) {
    const int e = tid + i * 256;
    yws[row * 2048 + e] = g[i] * scale * normw[e];
  }
}

// ---------------------------------------------------------------------------
extern "C" void kernel_launch(void* const* d_in, const int* in_sizes, int n_in,
                              void* d_out, int out_size, void* d_ws, size_t ws_size,
                              hipStream_t stream) {
  (void)in_sizes; (void)n_in; (void)out_size; (void)ws_size;
  const float* u       = (const float*)d_in[0];  // (2,2048,1024)
  const float* W_in    = (const float*)d_in[1];  // (4384,1024)
  const float* conv_w  = (const float*)d_in[2];  // (2304,4)
  const float* conv_b  = (const float*)d_in[3];  // (2304,)
  const float* dt_bias = (const float*)d_in[4];  // (32,)
  const float* A_log   = (const float*)d_in[5];  // (32,)
  const float* Dvec    = (const float*)d_in[6];  // (32,)
  const float* norm_w  = (const float*)d_in[7];  // (2048,)
  const float* W_out   = (const float*)d_in[8];  // (1024,2048)
  float* out = (float*)d_out;                    // (2,2048,1024)
  float* ws = (float*)d_ws;

  float* ZX   = ws;                   // 4096 x 4384  in_proj output
  float* XBC  = ZX + 17956864LL;      // 4096 x 2304  conv+silu output
  float* DT   = XBC + 9437184LL;      // 4096 x 32
  float* CUM  = DT + 131072LL;        // 4096 x 32
  float* CB   = CUM + 131072LL;       // 16 x 256 x 256
  float* S    = CB + 1048576LL;       // 16 x 32 x 128 x 64
  float* PREV = S + 4194304LL;        // 16 x 32 x 128 x 64
  float* Y    = PREV + 4194304LL;     // 4096 x 2048
  // total: 45,481,984 floats (~182 MB)

  // 1) in_proj: zxbcdt = u @ W_in^T   (M=4096, N=4384, K=1024)
  gemm_nt_f32<<<dim3(32, 137, 1), 256, 0, stream>>>(u, 1024, 0, W_in, 1024, 0,
                                                    ZX, 4384, 0, 1024);
  // 2) causal conv1d + silu
  conv_silu<<<36864, 256, 0, stream>>>(ZX, conv_w, conv_b, XBC);
  // 3) dt softplus + per-chunk cumsum(dA)
  dt_cumsum<<<2, 256, 0, stream>>>(ZX, dt_bias, A_log, DT, CUM);
  // 4) CB[l,s] = C @ B^T per (b,chunk)  (M=N=256, K=128, batch=16)
  gemm_nt_f32<<<dim3(2, 8, 16), 256, 0, stream>>>(
      XBC + 2176, 2304, 256LL * 2304, XBC + 2048, 2304, 256LL * 2304,
      CB, 256, 65536LL, 128);
  // 5) intra-chunk Y + chunk states S (+ D*x)
  ssd_intra<<<dim3(32, 8, 2), 256, 0, stream>>>(XBC, CB, CUM, DT, Dvec, Y, S);
  // 6) inter-chunk state recurrence
  ssd_scan<<<2048, 256, 0, stream>>>(S, CUM, PREV);
  // 7) Y += exp(cum) * C @ prev^T
  ssd_inter<<<dim3(32, 8, 2), 256, 0, stream>>>(XBC, PREV, CUM, Y);
  // 8) gated RMSNorm (in place on Y, gate z from ZX)
  gated_rmsnorm<<<4096, 256, 0, stream>>>(Y, ZX, norm_w);
  // 9) out_proj: out = Y @ W_out^T  (M=4096, N=1024, K=2048)
  gemm_nt_f32<<<dim3(32, 32, 1), 256, 0, stream>>>(Y, 2048, 0, W_out, 2048, 0,
                                                   out, 1024, 0, 2048);
}